// MultiHeadAttention_48198122995866
// MI455X (gfx1250) — compile-verified
//
#include <hip/hip_runtime.h>

typedef float v2f __attribute__((ext_vector_type(2)));
typedef float v8f __attribute__((ext_vector_type(8)));

#define B_SZ   4
#define T_SZ   2048
#define D_IN_  256
#define NHEAD  8
#define DK     32
#define SCALE_ 0.17677669529663687f   /* 1/sqrt(32) */
#define LN_EPS 1e-5f
#define TPAD   2052                   /* padded LDS row stride (floats): 2052%64=4 */

__device__ __forceinline__ v8f wmma_f32(v2f a, v2f b, v8f c) {
  // D = A(16x4) * B(4x16) + C, fp32, wave32
  return __builtin_amdgcn_wmma_f32_16x16x4_f32(
      false, a, false, b, (short)0, c, false, false);
}

// ---------------------------------------------------------------------------
// Kernel 1: P = X @ W^T   (M=B*T=8192, N=256, K=256), torch-style W[out,in].
// One wave per 16x16 output tile. blockIdx.z selects Q/K/V.  (~3.2 GF, tiny)
// ---------------------------------------------------------------------------
__global__ __launch_bounds__(32) void qkv_proj_kernel(
    const float* __restrict__ q_in, const float* __restrict__ k_in,
    const float* __restrict__ v_in,
    const float* __restrict__ Wq, const float* __restrict__ Wk,
    const float* __restrict__ Wv,
    float* __restrict__ Qo, float* __restrict__ Ko, float* __restrict__ Vo)
{
  const float* X; const float* W; float* O;
  if (blockIdx.z == 0)      { X = q_in; W = Wq; O = Qo; }
  else if (blockIdx.z == 1) { X = k_in; W = Wk; O = Ko; }
  else                      { X = v_in; W = Wv; O = Vo; }

  const int m0   = blockIdx.x * 16;
  const int n0   = blockIdx.y * 16;
  const int lane = threadIdx.x;
  const int half = lane >> 4;       // 0: K=0,1   1: K=2,3
  const int l16  = lane & 15;
  const int koff = half * 2;

  const float* ap = X + (size_t)(m0 + l16) * D_IN_ + koff;
  const float* bp = W + (size_t)(n0 + l16) * D_IN_ + koff;

  v8f acc = {};
  for (int kk = 0; kk < D_IN_; kk += 4) {
    v2f a; a.x = ap[kk]; a.y = ap[kk + 1];
    v2f b; b.x = bp[kk]; b.y = bp[kk + 1];
    acc = wmma_f32(a, b, acc);
  }

  const int row = m0 + half * 8;
  const int col = n0 + l16;
#pragma unroll
  for (int i = 0; i < 8; ++i)
    O[(size_t)(row + i) * D_IN_ + col] = acc[i];
}

// ---------------------------------------------------------------------------
// Kernel 2 (fused): per (h*B+b, 16-row l-stripe):
//   phase 0: async-copy prev[16 x 2048] stripe into LDS      (ASYNCcnt path)
//   phase 1: scores = Q K^T (WMMA f32) * SCALE + LDS(prev), RMW in LDS
//   phase 2: row softmax in LDS; write normalized attention to HBM (once)
//   phase 3: out_pre(16x32) = attn(LDS) @ V, K split over 8 waves + LDS reduce
// Dynamic LDS: 16*TPAD floats (stripe) + 8*512 floats (AV partials) ~147.7 KB.
// ---------------------------------------------------------------------------
__global__ __launch_bounds__(256) void attn_fused_kernel(
    const float* __restrict__ Q, const float* __restrict__ Km,
    const float* __restrict__ Vm, const float* __restrict__ prev,
    float* __restrict__ attn, float* __restrict__ out_pre)
{
  extern __shared__ float lds[];
  float* stripe = lds;                 // [16][TPAD]
  float* red    = lds + 16 * TPAD;     // [8][16][32]

  const int tid  = threadIdx.x;
  const int wave = tid >> 5;
  const int lane = tid & 31;
  const int half = lane >> 4, l16 = lane & 15, koff = half * 2;

  const int lt = blockIdx.x;           // l-tile 0..127
  const int hb = blockIdx.y;           // h*B + b
  const int b  = hb & 3, h = hb >> 2;
  const int l0 = lt * 16;

  const size_t attn_base = ((size_t)hb * T_SZ + l0) * T_SZ;

  // ---- Phase 0: LDS stripe <- prev (async global->LDS, 16B granules) ----
  for (int r = 0; r < 16; ++r) {
    const float* src = prev + attn_base + (size_t)r * T_SZ;
    float* dst = stripe + r * TPAD;
    for (int t4 = tid; t4 < T_SZ / 4; t4 += 256) {
      const float* g = src + t4 * 4;
      unsigned lo = (unsigned)(uintptr_t)(dst + t4 * 4);  // flat[31:0] = LDS addr
      asm volatile("global_load_async_to_lds_b128 %0, %1, off"
                   :: "v"(lo), "v"(g) : "memory");
    }
  }
  asm volatile("s_wait_asynccnt 0" ::: "memory");
  __syncthreads();

  // ---- Phase 1: scores. Q A-fragment loaded once, reused for 16 t-tiles ----
  v2f aq[8];
  const float* qp = Q + (size_t)(b * T_SZ + l0 + l16) * D_IN_ + h * DK + koff;
#pragma unroll
  for (int kk = 0; kk < 8; ++kk) { aq[kk].x = qp[kk * 4]; aq[kk].y = qp[kk * 4 + 1]; }

  for (int tt = wave; tt < T_SZ / 16; tt += 8) {
    const int t0 = tt * 16;
    const float* kp = Km + (size_t)(b * T_SZ + t0 + l16) * D_IN_ + h * DK + koff;
    v8f acc = {};
#pragma unroll
    for (int kk = 0; kk < 8; ++kk) {
      v2f bb; bb.x = kp[kk * 4]; bb.y = kp[kk * 4 + 1];
      acc = wmma_f32(aq[kk], bb, acc);
    }
    float* sp = stripe + (half * 8) * TPAD + t0 + l16;   // rows half*8+i, col t0+l16
#pragma unroll
    for (int i = 0; i < 8; ++i)
      sp[i * TPAD] = acc[i] * SCALE_ + sp[i * TPAD];
  }
  __syncthreads();

  // ---- Phase 2: softmax per row (2 rows per wave), write attention once ----
#pragma unroll
  for (int rr = 0; rr < 2; ++rr) {
    const int r = wave * 2 + rr;
    float* row = stripe + r * TPAD;
    float* gout = attn + attn_base + (size_t)r * T_SZ;

    float m = -3.402823466e38f;
    for (int t = lane; t < T_SZ; t += 32) m = fmaxf(m, row[t]);
#pragma unroll
    for (int s = 16; s > 0; s >>= 1) m = fmaxf(m, __shfl_xor(m, s, 32));

    float sum = 0.0f;
    for (int t = lane; t < T_SZ; t += 32) {
      float e = __expf(row[t] - m);
      row[t] = e;
      sum += e;
    }
#pragma unroll
    for (int s = 16; s > 0; s >>= 1) sum += __shfl_xor(sum, s, 32);

    const float inv = 1.0f / sum;
    for (int t = lane; t < T_SZ; t += 32) {
      float v = row[t] * inv;
      row[t]  = v;
      gout[t] = v;
    }
  }
  __syncthreads();

  // ---- Phase 3: out = attn(LDS) @ V; each wave takes K-range of 256 ----
  {
    const int k0 = wave * 256;
    const float* vp = Vm + (size_t)(b * T_SZ + k0 + koff) * D_IN_ + h * DK + l16;
    const float* ap = stripe + l16 * TPAD + k0 + koff;   // A[M=l16][k], bank-spread by TPAD
    v8f acc0 = {}, acc1 = {};
    for (int k = 0; k < 256; k += 4) {
      v2f a; a.x = ap[k]; a.y = ap[k + 1];
      const float* vrow = vp + (size_t)k * D_IN_;
      v2f b0; b0.x = vrow[0];  b0.y = vrow[D_IN_];
      v2f b1; b1.x = vrow[16]; b1.y = vrow[D_IN_ + 16];
      acc0 = wmma_f32(a, b0, acc0);
      acc1 = wmma_f32(a, b1, acc1);
    }
    float* rp = red + wave * 512 + (half * 8) * 32 + l16;
#pragma unroll
    for (int i = 0; i < 8; ++i) {
      rp[i * 32]      = acc0[i];
      rp[i * 32 + 16] = acc1[i];
    }
  }
  __syncthreads();

  // reduce 8 wave-partials -> global out_pre[b, l0+m, h*32+n]
  for (int idx = tid; idx < 512; idx += 256) {
    float s = 0.0f;
#pragma unroll
    for (int w = 0; w < 8; ++w) s += red[w * 512 + idx];
    const int m = idx >> 5, n = idx & 31;
    out_pre[(size_t)(b * T_SZ + l0 + m) * D_IN_ + h * DK + n] = s;
  }
}

// ---------------------------------------------------------------------------
// Kernel 3: LayerNorm(out_pre) * gamma + beta + residual. One block per row.
// ---------------------------------------------------------------------------
__global__ __launch_bounds__(256) void ln_kernel(
    const float* __restrict__ out_pre, const float* __restrict__ residual,
    const float* __restrict__ gamma, const float* __restrict__ beta,
    float* __restrict__ out)
{
  const size_t row = blockIdx.x;
  const int c = threadIdx.x;
  const float x = out_pre[row * D_IN_ + c];

  __shared__ float s1[256];
  __shared__ float s2[256];
  s1[c] = x; s2[c] = x * x; __syncthreads();
  for (int s = 128; s > 0; s >>= 1) {
    if (c < s) { s1[c] += s1[c + s]; s2[c] += s2[c + s]; }
    __syncthreads();
  }
  const float mu  = s1[0] * (1.0f / D_IN_);
  const float var = s2[0] * (1.0f / D_IN_) - mu * mu;
  const float y = (x - mu) * rsqrtf(var + LN_EPS) * gamma[c] + beta[c]
                + residual[row * D_IN_ + c];
  out[row * D_IN_ + c] = y;
}

// ---------------------------------------------------------------------------
extern "C" void kernel_launch(void* const* d_in, const int* in_sizes, int n_in,
                              void* d_out, int out_size, void* d_ws, size_t ws_size,
                              hipStream_t stream)
{
  (void)in_sizes; (void)n_in; (void)out_size; (void)ws_size;

  const float* q_in  = (const float*)d_in[0];
  const float* k_in  = (const float*)d_in[1];
  const float* v_in  = (const float*)d_in[2];
  const float* prev  = (const float*)d_in[3];
  const float* Wq    = (const float*)d_in[4];
  const float* Wk    = (const float*)d_in[5];
  const float* Wv    = (const float*)d_in[6];
  const float* gamma = (const float*)d_in[7];
  const float* beta  = (const float*)d_in[8];

  const size_t MAT = (size_t)B_SZ * T_SZ * D_IN_;   // 2,097,152 elements

  float* out  = (float*)d_out;          // (B,T,256)
  float* attn = out + MAT;              // (H,B,T,T) — 2nd tuple output

  float* ws      = (float*)d_ws;        // 32 MB scratch
  float* Q       = ws;
  float* K       = ws + MAT;
  float* V       = ws + 2 * MAT;
  float* out_pre = ws + 3 * MAT;

  // 1) Q/K/V projections (X @ W^T)
  qkv_proj_kernel<<<dim3(B_SZ * T_SZ / 16, D_IN_ / 16, 3), 32, 0, stream>>>(
      q_in, k_in, v_in, Wq, Wk, Wv, Q, K, V);

  // 2) fused scores + softmax + A@V
  const size_t LDS_BYTES = (size_t)(16 * TPAD + 8 * 512) * sizeof(float);
  attn_fused_kernel<<<dim3(T_SZ / 16, NHEAD * B_SZ), 256, LDS_BYTES, stream>>>(
      Q, K, V, prev, attn, out_pre);

  // 3) LayerNorm + residual
  ln_kernel<<<dim3(B_SZ * T_SZ), 256, 0, stream>>>(out_pre, q_in, gamma, beta, out);
}